// PhaseFunctionedNet_6279242186959
// MI455X (gfx1250) — compile-verified
//
#include <hip/hip_runtime.h>
#include <math.h>

// PFNN: per layer, y_b = sum_j d_j(b) * ((X @ W[j])_b + bias[j]), ELU between.
// Cubic weight interpolation is linear in the 4 phase banks, so it is hoisted
// out of the GEMM: 4 dense shared-weight GEMMs + per-row 4-way combine,
// computed with V_WMMA_F32_16X16X4_F32 in full f32 precision.

typedef __attribute__((ext_vector_type(2))) float v2f;
typedef __attribute__((ext_vector_type(8))) float v8f;

#define BATCH 512
#define IN_F  342
#define HID   512
#define OUT_F 311

// ---------------------------------------------------------------------------
// Per-sample Catmull-Rom coefficients scattered by phase index.
// ---------------------------------------------------------------------------
__global__ void pfnn_coeffs(const float* __restrict__ p, float* __restrict__ dcoef) {
    int b = blockIdx.x * blockDim.x + threadIdx.x;
    if (b >= BATCH) return;
    float p4  = p[b] * 4.0f;
    float mu  = p4 - floorf(p4);          // jnp.mod(p4, 1.0), p >= 0
    int   i1  = ((int)p4) & 3;            // p4.astype(int32) % 4
    float mu2 = mu * mu;
    float mu3 = mu2 * mu;
    float c0 = -0.5f * mu3 +        mu2 - 0.5f * mu;
    float c1 =  1.5f * mu3 - 2.5f * mu2 + 1.0f;
    float c2 = -1.5f * mu3 + 2.0f * mu2 + 0.5f * mu;
    float c3 =  0.5f * mu3 - 0.5f * mu2;
    float d[4];
    d[(i1 + 3) & 3] = c0;
    d[ i1         ] = c1;
    d[(i1 + 1) & 3] = c2;
    d[(i1 + 2) & 3] = c3;
    ((float4*)dcoef)[b] = make_float4(d[0], d[1], d[2], d[3]);
}

// ---------------------------------------------------------------------------
// One wave32 computes a 16x32 output tile (2 column sub-tiles, 4 phase banks).
//   A : [BATCH][K] row major      W : [4][K][N] row major
//   bias : [4][N]                 out : [BATCH][N] (only col < N written)
// Out-of-range columns: addresses clamped, garbage results discarded at store
// (a bad B column only pollutes the same D column). K-tail: A zeroed via
// branch-free select, B rows clamped to valid memory. EXEC stays all-ones.
// ---------------------------------------------------------------------------
template <int K, int N, bool ELU>
__launch_bounds__(32)
__global__ void pfnn_layer(const float* __restrict__ A,
                           const float* __restrict__ W,
                           const float* __restrict__ bias,
                           const float* __restrict__ dcoef,
                           float*       __restrict__ out) {
    constexpr int KMAIN = K & ~3;

    const int lane  = threadIdx.x;        // 0..31 (wave32)
    const int m16   = lane & 15;          // M for A, N for B/C
    const int kgrp  = lane >> 4;          // K pair group (0/1)
    const int tileM = blockIdx.y * 16;
    const int col0  = blockIdx.x * 32 + m16;
    const int col1  = col0 + 16;
    const int cs0   = (col0 < N) ? col0 : (N - 1);   // clamped (always in-bounds)
    const int cs1   = (col1 < N) ? col1 : (N - 1);

    v8f acc[4][2] = {};                   // [bank][column sub-tile]

    const float* Arow = A + (size_t)(tileM + m16) * K;
    const int    krow = kgrp * 2;

#pragma unroll 2
    for (int k = 0; k < KMAIN; k += 4) {
        const int kk0 = k + krow;         // even -> 8B aligned (K rows are even)
        v2f a = *(const v2f*)(Arow + kk0);
#pragma unroll
        for (int j = 0; j < 4; ++j) {
            const float* Wr0 = W + ((size_t)j * K + kk0) * N;  // row kk0
            const float* Wr1 = Wr0 + N;                        // row kk0+1
            v2f b0, b1;
            b0.x = Wr0[cs0]; b0.y = Wr1[cs0];
            b1.x = Wr0[cs1]; b1.y = Wr1[cs1];
            acc[j][0] = __builtin_amdgcn_wmma_f32_16x16x4_f32(
                false, a, false, b0, (short)0, acc[j][0], false, false);
            acc[j][1] = __builtin_amdgcn_wmma_f32_16x16x4_f32(
                false, a, false, b1, (short)0, acc[j][1], false, false);
        }
    }

    if constexpr ((K & 3) != 0) {         // peeled K tail (layer 0: K = 342)
        const int kk0 = KMAIN + krow;
        const int kk1 = kk0 + 1;
        const int ks0 = (kk0 < K) ? kk0 : (K - 1);   // clamped row indices
        const int ks1 = (kk1 < K) ? kk1 : (K - 1);
        v2f a;
        float a0 = Arow[ks0];
        float a1 = Arow[ks1];
        a.x = (kk0 < K) ? a0 : 0.0f;      // zeroing A alone kills the products
        a.y = (kk1 < K) ? a1 : 0.0f;
#pragma unroll
        for (int j = 0; j < 4; ++j) {
            const float* Wj = W + (size_t)j * K * N;
            v2f b0, b1;
            b0.x = Wj[(size_t)ks0 * N + cs0]; b0.y = Wj[(size_t)ks1 * N + cs0];
            b1.x = Wj[(size_t)ks0 * N + cs1]; b1.y = Wj[(size_t)ks1 * N + cs1];
            acc[j][0] = __builtin_amdgcn_wmma_f32_16x16x4_f32(
                false, a, false, b0, (short)0, acc[j][0], false, false);
            acc[j][1] = __builtin_amdgcn_wmma_f32_16x16x4_f32(
                false, a, false, b1, (short)0, acc[j][1], false, false);
        }
    }

    // --- epilogue: 4-bank combine + bias + optional ELU -------------------
    float bj0[4], bj1[4];
#pragma unroll
    for (int j = 0; j < 4; ++j) {         // hoisted, clamped bias loads
        bj0[j] = bias[j * N + cs0];
        bj1[j] = bias[j * N + cs1];
    }

    const float4* dmat   = (const float4*)dcoef;
    const int     rowHalf = kgrp * 8;     // C/D: VGPR v -> M = v + 8*(lane>=16)
#pragma unroll
    for (int v = 0; v < 8; ++v) {
        const int row = tileM + v + rowHalf;
        const float4 dv = dmat[row];
        float e0 = dv.x * (acc[0][0][v] + bj0[0]) + dv.y * (acc[1][0][v] + bj0[1])
                 + dv.z * (acc[2][0][v] + bj0[2]) + dv.w * (acc[3][0][v] + bj0[3]);
        float e1 = dv.x * (acc[0][1][v] + bj1[0]) + dv.y * (acc[1][1][v] + bj1[1])
                 + dv.z * (acc[2][1][v] + bj1[2]) + dv.w * (acc[3][1][v] + bj1[3]);
        if (ELU) {
            e0 = (e0 > 0.0f) ? e0 : (expf(e0) - 1.0f);
            e1 = (e1 > 0.0f) ? e1 : (expf(e1) - 1.0f);
        }
        if (col0 < N) out[(size_t)row * N + col0] = e0;
        if (col1 < N) out[(size_t)row * N + col1] = e1;
    }
}

// ---------------------------------------------------------------------------
extern "C" void kernel_launch(void* const* d_in, const int* in_sizes, int n_in,
                              void* d_out, int out_size, void* d_ws, size_t ws_size,
                              hipStream_t stream) {
    const float* p  = (const float*)d_in[0];  // [512]
    const float* x  = (const float*)d_in[1];  // [512][342]
    const float* W0 = (const float*)d_in[2];  // [4][342][512]
    const float* b0 = (const float*)d_in[3];  // [4][512]
    const float* W1 = (const float*)d_in[4];  // [4][512][512]
    const float* b1 = (const float*)d_in[5];  // [4][512]
    const float* W2 = (const float*)d_in[6];  // [4][512][311]
    const float* b2 = (const float*)d_in[7];  // [4][311]
    float* y = (float*)d_out;                 // [512][311]

    char*  ws    = (char*)d_ws;
    float* dcoef = (float*)ws;                                    // 512*4 f32
    float* h0    = (float*)(ws + 8192);                           // 512*512 f32
    float* h1    = (float*)(ws + 8192 + (size_t)BATCH * HID * 4); // 512*512 f32

    pfnn_coeffs<<<dim3((BATCH + 127) / 128), dim3(128), 0, stream>>>(p, dcoef);

    const dim3 blk(32, 1, 1);  // one wave32 per block: EXEC all-ones at WMMA
    pfnn_layer<IN_F, HID,  true ><<<dim3((HID   + 31) / 32, BATCH / 16), blk, 0, stream>>>(
        x,  W0, b0, dcoef, h0);
    pfnn_layer<HID,  HID,  true ><<<dim3((HID   + 31) / 32, BATCH / 16), blk, 0, stream>>>(
        h0, W1, b1, dcoef, h1);
    pfnn_layer<HID,  OUT_F,false><<<dim3((OUT_F + 31) / 32, BATCH / 16), blk, 0, stream>>>(
        h1, W2, b2, dcoef, y);
}